// MeanAggregator_88974542504021
// MI455X (gfx1250) — compile-verified
//
#include <hip/hip_runtime.h>

typedef float v2f __attribute__((ext_vector_type(2)));
typedef float v8f __attribute__((ext_vector_type(8)));

#define FEAT 64  // feature dim of reference (x is [N, 64] f32)

// ---------------------------------------------------------------------------
// Pass 0: zero the sum buffer (d_out) and the per-node counts (d_ws).
// ---------------------------------------------------------------------------
__global__ void zero_kernel(float* __restrict__ sums, float* __restrict__ cnt,
                            int nSums, int nCnt) {
    int stride = gridDim.x * blockDim.x;
    for (int i = blockIdx.x * blockDim.x + threadIdx.x; i < nSums; i += stride)
        sums[i] = 0.0f;
    for (int i = blockIdx.x * blockDim.x + threadIdx.x; i < nCnt; i += stride)
        cnt[i] = 0.0f;
}

// ---------------------------------------------------------------------------
// Pass 1: one wave (32 lanes) per edge. Lane l owns features [2l, 2l+1].
// Gather x[row] as a coalesced 256B row (float2 per lane), scatter with
// hardware f32 atomics (unsafeAtomicAdd -> native GLOBAL_ATOMIC_ADD_F32,
// no CAS loop). All atomic targets (12.8MB) stay resident in the 192MB L2.
// ---------------------------------------------------------------------------
__global__ void scatter_kernel(const float* __restrict__ x,
                               const long long* __restrict__ eidx,
                               float* __restrict__ sums,
                               float* __restrict__ cnt,
                               int E) {
    int t    = blockIdx.x * blockDim.x + threadIdx.x;
    int e    = t >> 5;
    int lane = t & 31;
    if (e >= E) return;

    long long row = eidx[e];       // edge_index[0][e]
    long long col = eidx[E + e];   // edge_index[1][e]

    const float2 v = *(const float2*)(x + row * FEAT + lane * 2);
    float* dst = sums + col * FEAT + lane * 2;
    unsafeAtomicAdd(dst,     v.x);
    unsafeAtomicAdd(dst + 1, v.y);
    if (lane == 0)
        unsafeAtomicAdd(cnt + col, 1.0f);
}

// ---------------------------------------------------------------------------
// Pass 2: out = diag(1/max(count,1)) * sums, one 16x16 tile per wave via
// four chained V_WMMA_F32_16X16X4_F32 (K=16 split into 4 slices of the
// diagonal). Layouts per CDNA5 ISA:
//   A 16x4 f32 : lanes 0-15 -> M=lane, K={0,1}; lanes 16-31 -> M=lane-16, K={2,3}
//   B 4x16 f32 : lanes 0-15 -> N=lane, K={0,1}; lanes 16-31 -> N=lane-16, K={2,3}
//   C/D 16x16  : VGPR r -> row r + 8*(lane>>4), column = lane&15
// Tile-id early exit is wave-uniform, so EXEC is all-ones at the WMMAs.
// ---------------------------------------------------------------------------
__global__ void finalize_wmma_kernel(float* __restrict__ sums,
                                     const float* __restrict__ cnt,
                                     int numTiles) {
    int lane = threadIdx.x & 31;
    int tile = blockIdx.x * (blockDim.x >> 5) + (threadIdx.x >> 5);
    if (tile >= numTiles) return;

    const int node_base = (tile >> 2) << 4;   // 16 nodes per tile
    const int feat_base = (tile & 3) << 4;    // 4 feature tiles of 16
    const int m     = lane & 15;              // row (A) / column (B, C/D)
    const int khalf = lane >> 4;
    const int k0    = khalf * 2;
    const int k1    = k0 + 1;

    const float c = 1.0f / fmaxf(cnt[node_base + m], 1.0f);
    const float* S = sums + (long long)node_base * FEAT + feat_base;

    v8f acc = {};
#pragma unroll
    for (int j = 0; j < 4; ++j) {
        v2f a, b;
        a.x = (m == 4 * j + k0) ? c : 0.0f;   // diag(c) columns 4j..4j+3
        a.y = (m == 4 * j + k1) ? c : 0.0f;
        b.x = S[(4 * j + k0) * FEAT + m];     // sums rows 4j..4j+3
        b.y = S[(4 * j + k1) * FEAT + m];
        acc = __builtin_amdgcn_wmma_f32_16x16x4_f32(
            /*neg_a=*/false, a, /*neg_b=*/false, b,
            /*c_mod=*/(short)0, acc, /*reuse_a=*/false, /*reuse_b=*/false);
    }

#pragma unroll
    for (int r = 0; r < 8; ++r)
        sums[(long long)(node_base + r + 8 * khalf) * FEAT + feat_base + m] = acc[r];
}

// Scalar fallback for a (non-existent here: 50000 % 16 == 0) node tail.
__global__ void finalize_tail_kernel(float* __restrict__ sums,
                                     const float* __restrict__ cnt,
                                     int firstNode, int N) {
    int i = blockIdx.x * blockDim.x + threadIdx.x;
    int node = firstNode + (i >> 6);
    int f    = i & 63;
    if (node >= N) return;
    sums[(long long)node * FEAT + f] /= fmaxf(cnt[node], 1.0f);
}

extern "C" void kernel_launch(void* const* d_in, const int* in_sizes, int n_in,
                              void* d_out, int out_size, void* d_ws, size_t ws_size,
                              hipStream_t stream) {
    const float*     x    = (const float*)d_in[0];      // [N, 64] f32
    const long long* eidx = (const long long*)d_in[1];  // [2, E] i64, row-major
    float* sums = (float*)d_out;                        // [N, 64] f32 (also output)
    float* cnt  = (float*)d_ws;                         // [N] f32 counts

    const int ND = in_sizes[0];
    const int N  = ND / FEAT;
    const int E  = in_sizes[1] / 2;

    zero_kernel<<<1024, 256, 0, stream>>>(sums, cnt, ND, N);

    long long totalT = (long long)E * 32;
    int sblocks = (int)((totalT + 255) / 256);
    scatter_kernel<<<sblocks, 256, 0, stream>>>(x, eidx, sums, cnt, E);

    const int fullNodeGroups = N / 16;
    const int numTiles = fullNodeGroups * 4;           // 12500 waves for N=50000
    if (numTiles > 0) {
        int fblocks = (numTiles + 7) / 8;              // 8 waves per 256-thr block
        finalize_wmma_kernel<<<fblocks, 256, 0, stream>>>(sums, cnt, numTiles);
    }
    const int firstTail = fullNodeGroups * 16;
    if (firstTail < N) {
        int tailElems = (N - firstTail) * FEAT;
        finalize_tail_kernel<<<(tailElems + 255) / 256, 256, 0, stream>>>(
            sums, cnt, firstTail, N);
    }
}